// GATLayer_53455162966092
// MI455X (gfx1250) — compile-verified
//
#include <hip/hip_runtime.h>

typedef __attribute__((ext_vector_type(2))) float v2f;
typedef __attribute__((ext_vector_type(8))) float v8f;

#define N_ITEM 100000
#define N_TAG  50000
#define HD     256            // H*D = 4*64
#define E_I2T  1000000
#define E_T2T  500000
#define NEG_INF (-__builtin_huge_valf())

__device__ __forceinline__ float lrelu(float x) { return x > 0.f ? x : 0.2f * x; }

// reduce across the 16 lanes of a half-wave, one atomic per half-wave
__device__ __forceinline__ void redAtomic16(float p, int lane, float* addr) {
    p += __shfl_xor(p, 1, 32);
    p += __shfl_xor(p, 2, 32);
    p += __shfl_xor(p, 4, 32);
    p += __shfl_xor(p, 8, 32);
    if ((lane & 15) == 0) atomicAdd(addr, p);
}

// one-time weight transpose: WT[n*128+k] = W[k*256+n]  (makes B fragments contiguous v2f)
__global__ __launch_bounds__(256) void transpose_w_kernel(
    const float* __restrict__ W, float* __restrict__ WT)
{
    int i = blockIdx.x * 256 + threadIdx.x;     // over 128*256
    int k = i >> 8, n = i & 255;
    WT[n * 128 + k] = W[i];
}

// C = A[M,128] @ W[128,256] via V_WMMA_F32_16X16X4_F32, W pre-transposed (WT[256,128]).
// Block = 128 threads = 4 waves; wave w owns rows [row0,row0+16) x cols [w*64,w*64+64).
// Fused epilogue: optional store of H, optional attn_l/attn_r dot products -> el/er.
__global__ __launch_bounds__(128) void gemm_attn_kernel(
    const float* __restrict__ A, const float* __restrict__ WT,
    float* __restrict__ Hout,
    const float* __restrict__ attnL, float* __restrict__ el,
    const float* __restrict__ attnR, float* __restrict__ er)
{
    const int lane = threadIdx.x & 31;
    const int wave = threadIdx.x >> 5;
    const int row0 = blockIdx.x << 4;
    const int n0   = wave << 6;
    const int mrow = lane & 15;
    const int koff = (lane >> 4) << 1;   // lanes 16-31 carry K+2,K+3

    v8f acc[4] = {};

    const float* arow = A  + (size_t)(row0 + mrow) * 128 + koff;
    const float* wt   = WT + (size_t)(n0 + mrow) * 128 + koff;

    for (int k = 0; k < 128; k += 4) {
        v2f a  = *(const v2f*)(arow + k);
        v2f b0 = *(const v2f*)(wt + k);
        v2f b1 = *(const v2f*)(wt + 16 * 128 + k);
        v2f b2 = *(const v2f*)(wt + 32 * 128 + k);
        v2f b3 = *(const v2f*)(wt + 48 * 128 + k);
        acc[0] = __builtin_amdgcn_wmma_f32_16x16x4_f32(false, a, false, b0, (short)0, acc[0], false, false);
        acc[1] = __builtin_amdgcn_wmma_f32_16x16x4_f32(false, a, false, b1, (short)0, acc[1], false, false);
        acc[2] = __builtin_amdgcn_wmma_f32_16x16x4_f32(false, a, false, b2, (short)0, acc[2], false, false);
        acc[3] = __builtin_amdgcn_wmma_f32_16x16x4_f32(false, a, false, b3, (short)0, acc[3], false, false);
    }

    const int half8 = (lane >> 4) << 3;  // upper half-wave holds rows M+8..M+15
#pragma unroll
    for (int t = 0; t < 4; ++t) {
        const int col = n0 + (t << 4) + (lane & 15);
        const int h   = col >> 6;        // a 16-col tile never crosses a 64-wide head
        const float aL = attnL ? attnL[col] : 0.f;
        const float aR = attnR ? attnR[col] : 0.f;
#pragma unroll
        for (int v = 0; v < 8; ++v) {
            const int row = row0 + half8 + v;
            const float val = acc[t][v];
            if (Hout)  Hout[(size_t)row * HD + col] = val;
            if (attnL) redAtomic16(val * aL, lane, &el[row * 4 + h]);
            if (attnR) redAtomic16(val * aR, lane, &er[row * 4 + h]);
        }
    }
}

// ---------------- CSR build (count -> scan -> scatter) ----------------

__global__ __launch_bounds__(256) void edge_count_kernel(
    const int* __restrict__ dst, int* __restrict__ cnt, int E)
{
    int e = blockIdx.x * 256 + threadIdx.x;
    if (e < E) atomicAdd(&cnt[dst[e]], 1);
}

// single-block exclusive scan over n counts -> base[0..n] (base[n] = total)
__global__ __launch_bounds__(1024) void scan_kernel(
    const int* __restrict__ cnt, int* __restrict__ base, int n)
{
    __shared__ int sm[1024];
    __shared__ int carrySm;
    const int tid = threadIdx.x;
    if (tid == 0) carrySm = 0;
    __syncthreads();
    for (int start = 0; start < n; start += 1024) {
        const int i = start + tid;
        const int v = (i < n) ? cnt[i] : 0;
        sm[tid] = v;
        __syncthreads();
        for (int o = 1; o < 1024; o <<= 1) {
            int t = (tid >= o) ? sm[tid - o] : 0;
            __syncthreads();
            sm[tid] += t;
            __syncthreads();
        }
        const int incl  = sm[tid];
        const int carry = carrySm;
        if (i < n) base[i] = carry + incl - v;   // exclusive
        __syncthreads();
        if (tid == 1023) carrySm = carry + incl;
        __syncthreads();
    }
    if (tid == 0) base[n] = carrySm;
}

__global__ __launch_bounds__(256) void edge_scatter_kernel(
    const int* __restrict__ src, const int* __restrict__ dst,
    const int* __restrict__ base, int* __restrict__ cursor,
    int* __restrict__ slots, int E)
{
    int e = blockIdx.x * 256 + threadIdx.x;
    if (e >= E) return;
    int d = dst[e];
    int pos = base[d] + atomicAdd(&cursor[d], 1);
    slots[pos] = src[e];
}

// ---------------- dst-centric softmax + message gather ----------------
// One wave per destination node: register-resident segment-max, exp-sum and
// weighted accumulation of hs[src]; single write of the 256-float output row.
// bias1 != null  -> write mode:  out = acc + bias1 + bias2
// bias1 == null  -> accumulate:  out += acc
__global__ __launch_bounds__(256) void dst_msg_kernel(
    const int* __restrict__ rowptr, const int* __restrict__ slots,
    const float* __restrict__ el, const float* __restrict__ er,
    const float* __restrict__ hs, float* __restrict__ out,
    const float* __restrict__ bias1, const float* __restrict__ bias2)
{
    const int lane = threadIdx.x & 31;
    const int d = (blockIdx.x << 3) + (threadIdx.x >> 5);   // 8 dsts per block
    const int j0 = rowptr[d], j1 = rowptr[d + 1];
    const float4 r = *(const float4*)(er + (size_t)d * 4);
    const int c0 = lane << 2;        // 4 contiguous cols, single head
    const int c1 = 128 + c0;
    const int hi = lane >> 4;        // 0: heads {0,2}; 1: heads {1,3}
    float4 acc0 = {0.f, 0.f, 0.f, 0.f};
    float4 acc1 = {0.f, 0.f, 0.f, 0.f};
    float* orow = out + (size_t)d * HD;

    if (j1 > j0) {
        // pass 1: per-head max of leaky-relu scores (tiny: 20B/edge)
        float mx0 = NEG_INF, mx1 = NEG_INF, mx2 = NEG_INF, mx3 = NEG_INF;
        for (int j = j0; j < j1; ++j) {
            int s = slots[j];
            float4 l = *(const float4*)(el + (size_t)s * 4);
            mx0 = fmaxf(mx0, lrelu(l.x + r.x));
            mx1 = fmaxf(mx1, lrelu(l.y + r.y));
            mx2 = fmaxf(mx2, lrelu(l.z + r.z));
            mx3 = fmaxf(mx3, lrelu(l.w + r.w));
        }
        // pass 2: accumulate exp-weights and weighted feature rows
        float s0 = 0.f, s1 = 0.f, s2 = 0.f, s3 = 0.f;
        int s = slots[j0];
        for (int j = j0; j < j1; ++j) {
            const int sn = (j + 1 < j1) ? slots[j + 1] : s;
            __builtin_prefetch(hs + (size_t)sn * HD + c0, 0, 0);  // global_prefetch_b8
            float4 l = *(const float4*)(el + (size_t)s * 4);
            float e0 = __expf(lrelu(l.x + r.x) - mx0);
            float e1 = __expf(lrelu(l.y + r.y) - mx1);
            float e2 = __expf(lrelu(l.z + r.z) - mx2);
            float e3 = __expf(lrelu(l.w + r.w) - mx3);
            s0 += e0; s1 += e1; s2 += e2; s3 += e3;
            const float* hrow = hs + (size_t)s * HD;
            float4 h0 = *(const float4*)(hrow + c0);
            float4 h1 = *(const float4*)(hrow + c1);
            const float a0 = hi ? e1 : e0;
            const float a1 = hi ? e3 : e2;
            acc0.x += h0.x * a0; acc0.y += h0.y * a0; acc0.z += h0.z * a0; acc0.w += h0.w * a0;
            acc1.x += h1.x * a1; acc1.y += h1.y * a1; acc1.z += h1.z * a1; acc1.w += h1.w * a1;
            s = sn;
        }
        const float inv0 = 1.f / (hi ? s1 : s0);
        const float inv1 = 1.f / (hi ? s3 : s2);
        acc0.x *= inv0; acc0.y *= inv0; acc0.z *= inv0; acc0.w *= inv0;
        acc1.x *= inv1; acc1.y *= inv1; acc1.z *= inv1; acc1.w *= inv1;
    }

    if (bias1) {
        float4 ba = *(const float4*)(bias1 + c0);
        float4 bb = *(const float4*)(bias2 + c0);
        float4 ca = *(const float4*)(bias1 + c1);
        float4 cb = *(const float4*)(bias2 + c1);
        acc0.x += ba.x + bb.x; acc0.y += ba.y + bb.y; acc0.z += ba.z + bb.z; acc0.w += ba.w + bb.w;
        acc1.x += ca.x + cb.x; acc1.y += ca.y + cb.y; acc1.z += ca.z + cb.z; acc1.w += ca.w + cb.w;
    } else {
        float4 o0 = *(const float4*)(orow + c0);
        float4 o1 = *(const float4*)(orow + c1);
        acc0.x += o0.x; acc0.y += o0.y; acc0.z += o0.z; acc0.w += o0.w;
        acc1.x += o1.x; acc1.y += o1.y; acc1.z += o1.z; acc1.w += o1.w;
    }
    *(float4*)(orow + c0) = acc0;
    *(float4*)(orow + c1) = acc1;
}

__global__ __launch_bounds__(256) void fill_kernel(float* __restrict__ p, float v, int n) {
    int i = blockIdx.x * 256 + threadIdx.x;
    if (i < n) p[i] = v;
}
__global__ __launch_bounds__(256) void fill_int_kernel(int* __restrict__ p, int v, int n) {
    int i = blockIdx.x * 256 + threadIdx.x;
    if (i < n) p[i] = v;
}

extern "C" void kernel_launch(void* const* d_in, const int* in_sizes, int n_in,
                              void* d_out, int out_size, void* d_ws, size_t ws_size,
                              hipStream_t stream)
{
    (void)in_sizes; (void)n_in; (void)out_size; (void)ws_size;
    const float* feat_item = (const float*)d_in[0];
    const float* feat_tag  = (const float*)d_in[1];
    const float* W_i2t     = (const float*)d_in[2];
    const float* al_i2t    = (const float*)d_in[3];
    const float* ar_i2t    = (const float*)d_in[4];
    const float* b_i2t     = (const float*)d_in[5];
    const float* W_t2t     = (const float*)d_in[6];
    const float* al_t2t    = (const float*)d_in[7];
    const float* ar_t2t    = (const float*)d_in[8];
    const float* b_t2t     = (const float*)d_in[9];
    const int* src_i2t = (const int*)d_in[10];
    const int* dst_i2t = (const int*)d_in[11];
    const int* src_t2t = (const int*)d_in[12];
    const int* dst_t2t = (const int*)d_in[13];
    float* out = (float*)d_out;

    char* p = (char*)d_ws;
    float* hs_item  = (float*)p; p += (size_t)N_ITEM * HD * 4;   // 102.4 MB
    float* hs_tag   = (float*)p; p += (size_t)N_TAG  * HD * 4;   //  51.2 MB
    float* el_item  = (float*)p; p += (size_t)N_ITEM * 4 * 4;    // contiguous zero block start
    float* er_i2t   = (float*)p; p += (size_t)N_TAG  * 4 * 4;
    float* el_t2t   = (float*)p; p += (size_t)N_TAG  * 4 * 4;
    float* er_t2t   = (float*)p; p += (size_t)N_TAG  * 4 * 4;
    float* WT_i2t   = (float*)p; p += (size_t)128 * 256 * 4;
    float* WT_t2t   = (float*)p; p += (size_t)128 * 256 * 4;
    int*   cnt_i2t  = (int*)p;   p += (size_t)N_TAG * 4;         // cnt arrays contiguous
    int*   cnt_t2t  = (int*)p;   p += (size_t)N_TAG * 4;
    int*   base_i2t = (int*)p;   p += (size_t)(N_TAG + 4) * 4;   // padded for 16B alignment
    int*   base_t2t = (int*)p;   p += (size_t)(N_TAG + 4) * 4;
    int*   slots_i2t = (int*)p;  p += (size_t)E_I2T * 4;
    int*   slots_t2t = (int*)p;  p += (size_t)E_T2T * 4;

    // init accumulators
    const int nzf = (N_ITEM + 3 * N_TAG) * 4;     // el_item..er_t2t
    fill_kernel<<<(nzf + 255) / 256, 256, 0, stream>>>(el_item, 0.f, nzf);
    fill_int_kernel<<<(2 * N_TAG + 255) / 256, 256, 0, stream>>>(cnt_i2t, 0, 2 * N_TAG);

    // weight transposes (once per call, 128 KB each)
    transpose_w_kernel<<<128, 256, 0, stream>>>(W_i2t, WT_i2t);
    transpose_w_kernel<<<128, 256, 0, stream>>>(W_t2t, WT_t2t);

    // GEMMs + fused attn projections (WMMA f32 16x16x4)
    gemm_attn_kernel<<<N_ITEM / 16, 128, 0, stream>>>(
        feat_item, WT_i2t, hs_item, al_i2t, el_item, nullptr, nullptr);
    gemm_attn_kernel<<<N_TAG / 16, 128, 0, stream>>>(
        feat_tag, WT_i2t, nullptr, nullptr, nullptr, ar_i2t, er_i2t);   // er only, no store
    gemm_attn_kernel<<<N_TAG / 16, 128, 0, stream>>>(
        feat_tag, WT_t2t, hs_tag, al_t2t, el_t2t, ar_t2t, er_t2t);

    // CSR build for both graphs (count -> scan -> scatter)
    edge_count_kernel<<<(E_I2T + 255) / 256, 256, 0, stream>>>(dst_i2t, cnt_i2t, E_I2T);
    edge_count_kernel<<<(E_T2T + 255) / 256, 256, 0, stream>>>(dst_t2t, cnt_t2t, E_T2T);
    scan_kernel<<<1, 1024, 0, stream>>>(cnt_i2t, base_i2t, N_TAG);
    scan_kernel<<<1, 1024, 0, stream>>>(cnt_t2t, base_t2t, N_TAG);
    fill_int_kernel<<<(2 * N_TAG + 255) / 256, 256, 0, stream>>>(cnt_i2t, 0, 2 * N_TAG); // reset cursors
    edge_scatter_kernel<<<(E_I2T + 255) / 256, 256, 0, stream>>>(
        src_i2t, dst_i2t, base_i2t, cnt_i2t, slots_i2t, E_I2T);
    edge_scatter_kernel<<<(E_T2T + 255) / 256, 256, 0, stream>>>(
        src_t2t, dst_t2t, base_t2t, cnt_t2t, slots_t2t, E_T2T);

    // dst-centric softmax + gather-accumulate (no atomics, one wave per dst)
    dst_msg_kernel<<<N_TAG / 8, 256, 0, stream>>>(
        base_i2t, slots_i2t, el_item, er_i2t, hs_item, out, b_i2t, b_t2t);   // write + bias
    dst_msg_kernel<<<N_TAG / 8, 256, 0, stream>>>(
        base_t2t, slots_t2t, el_t2t, er_t2t, hs_tag, out, nullptr, nullptr); // accumulate
}